// KvPageCache_60567628808533
// MI455X (gfx1250) — compile-verified
//
#include <hip/hip_runtime.h>
#include <hip/hip_bf16.h>
#include <stdint.h>

// native vector type so __builtin_nontemporal_load/store accept it
typedef __attribute__((ext_vector_type(4))) float v4f;

// Geometry fixed by the reference
constexpr int S  = 16;   // slots per page
constexpr int H  = 8;    // kv heads
constexpr int D  = 128;  // head dim
constexpr int HD4   = H * D / 4;      // 256 float4 per token per tensor (k or v)
constexpr int TILE4 = 2 * H * D / 4;  // 512 float4 per (page,slot) tile

constexpr int CPY_THREADS = 256;
constexpr int CPY_UNROLL  = 8;   // 16 B * 256 lanes * 8 = 32 KB per block

// GV-mode async ops add INST_OFFSET to BOTH the LDS address (VDST vgpr) and
// the global address (VADDR vgpr pair), so one base register pair per
// direction serves the whole unroll; the per-step 4 KB stride lives in the
// 24-bit offset immediate. NT hints keep the 256 MB single-touch stream from
// thrashing L2.
#define KV_ASYNC_LD(OFF)                                                       \
  asm volatile("global_load_async_to_lds_b128 %0, %1, off offset:" #OFF        \
               " th:TH_LOAD_NT"                                                \
               :: "v"(lds_addr), "v"(gsrc) : "memory")
#define KV_ASYNC_ST(OFF)                                                       \
  asm volatile("global_store_async_from_lds_b128 %0, %1, off offset:" #OFF     \
               " th:TH_STORE_NT"                                               \
               :: "v"(gdst), "v"(lds_addr) : "memory")

// ---------------------------------------------------------------------------
// Kernel 1: bulk copy via the CDNA5 async LDS DMA path.
// Per thread: 8x global_load_async_to_lds_b128 (global -> LDS, no data VGPRs),
// s_wait_asynccnt 0, 8x global_store_async_from_lds_b128 (LDS -> global).
// Each (tid, u) owns a unique 16-byte LDS slot; same-wave wait => no barrier.
// Grid covers the array exactly (8192 blocks x 32 KB = 256 MB).
// ---------------------------------------------------------------------------
__global__ __launch_bounds__(CPY_THREADS) void kv_copy_async_kernel(
    const char* __restrict__ src, char* __restrict__ dst) {
  __shared__ char buf[CPY_THREADS * 16 * CPY_UNROLL];

  const unsigned lds_addr =
      (unsigned)(reinterpret_cast<uintptr_t>(&buf[0]) & 0xffffffffu) +
      threadIdx.x * 16u;
  const size_t gbase =
      ((size_t)blockIdx.x * CPY_THREADS * CPY_UNROLL + threadIdx.x) * 16u;
  const char* gsrc = src + gbase;
  char*       gdst = dst + gbase;

  KV_ASYNC_LD(0);
  KV_ASYNC_LD(4096);
  KV_ASYNC_LD(8192);
  KV_ASYNC_LD(12288);
  KV_ASYNC_LD(16384);
  KV_ASYNC_LD(20480);
  KV_ASYNC_LD(24576);
  KV_ASYNC_LD(28672);

  asm volatile("s_wait_asynccnt 0x0" ::: "memory");

  KV_ASYNC_ST(0);
  KV_ASYNC_ST(4096);
  KV_ASYNC_ST(8192);
  KV_ASYNC_ST(12288);
  KV_ASYNC_ST(16384);
  KV_ASYNC_ST(20480);
  KV_ASYNC_ST(24576);
  KV_ASYNC_ST(28672);

  asm volatile("s_wait_asynccnt 0x0" ::: "memory");
}

// ---------------------------------------------------------------------------
// Tail copy (generality only; remainder is 0 for this problem's sizes).
// ---------------------------------------------------------------------------
__global__ __launch_bounds__(256) void kv_copy_tail_kernel(
    const v4f* __restrict__ src, v4f* __restrict__ dst, size_t start, size_t n4) {
  size_t i = start + (size_t)blockIdx.x * blockDim.x + threadIdx.x;
  if (i < n4) {
    v4f v = __builtin_nontemporal_load(&src[i]);
    __builtin_nontemporal_store(v, &dst[i]);
  }
}

// ---------------------------------------------------------------------------
// Kernel 2: scatter. One block per token; tokens t >= K are dropped.
//   out[page,slot, 2h  , :] = new_k[t, h, :]
//   out[page,slot, 2h+1, :] = new_v[t, h, :]
// Unique destinations => no ordering hazards within this kernel. Runs after
// the copy via stream ordering.
// ---------------------------------------------------------------------------
__global__ __launch_bounds__(256) void kv_scatter_kernel(
    const int* __restrict__ t_pages, const int* __restrict__ t_slots,
    const v4f* __restrict__ new_k, const v4f* __restrict__ new_v,
    const int* __restrict__ Kp, v4f* __restrict__ out) {
  const int t = blockIdx.x;
  if (t >= Kp[0]) return;  // fori_loop(0, K) prefix semantics

  const int page = t_pages[t];
  const int slot = t_slots[t];
  const size_t dst = ((size_t)page * S + slot) * TILE4;  // float4 units

  const v4f* kb = new_k + (size_t)t * HD4;
  const v4f* vb = new_v + (size_t)t * HD4;

  // 512 float4 per tile, 256 threads -> 2 per thread.
#pragma unroll
  for (int f = threadIdx.x; f < TILE4; f += 256) {
    const int h  = f >> 5;   // interleaved head 0..15
    const int d4 = f & 31;
    const v4f* row = (h & 1) ? vb : kb;
    v4f val = row[(size_t)(h >> 1) * 32 + d4];
    __builtin_nontemporal_store(val, &out[dst + f]);
  }
}

// ---------------------------------------------------------------------------
extern "C" void kernel_launch(void* const* d_in, const int* in_sizes, int n_in,
                              void* d_out, int out_size, void* d_ws, size_t ws_size,
                              hipStream_t stream) {
  const float* kv_pages = (const float*)d_in[0];
  const int*   t_pages  = (const int*)d_in[1];
  const int*   t_slots  = (const int*)d_in[2];
  const float* new_k    = (const float*)d_in[3];
  const float* new_v    = (const float*)d_in[4];
  const int*   Kp       = (const int*)d_in[5];  // device-resident scalar
  float*       out      = (float*)d_out;

  // ---- copy 256 MB kv_pages -> out via async LDS DMA ----
  const size_t n4 = (size_t)out_size / 4;            // float4 count
  const size_t per_block4 = (size_t)CPY_THREADS * CPY_UNROLL;  // 2048 float4
  const size_t full_blocks = n4 / per_block4;        // 8192 for this problem
  const size_t rem4 = n4 % per_block4;               // 0 for this problem

  if (full_blocks) {
    kv_copy_async_kernel<<<(int)full_blocks, CPY_THREADS, 0, stream>>>(
        (const char*)kv_pages, (char*)out);
  }
  if (rem4) {
    const int tb = (int)((rem4 + 255) / 256);
    kv_copy_tail_kernel<<<tb, 256, 0, stream>>>(
        (const v4f*)kv_pages, (v4f*)out, full_blocks * per_block4, n4);
  }

  // ---- scatter K tokens into their (page, slot) tiles ----
  const int T = in_sizes[1];  // number of tokens (2048)
  kv_scatter_kernel<<<T, 256, 0, stream>>>(
      t_pages, t_slots, (const v4f*)new_k, (const v4f*)new_v, Kp, (v4f*)out);
}